// OleLoss_2482491097949
// MI455X (gfx1250) — compile-verified
//
#include <hip/hip_runtime.h>
#include <hip/hip_bf16.h>
#include <math.h>

#define NN 8192
#define DD 128
#define CC 64
#define NS_ITERS 22
#define LAMBDA_ 0.25f
#define DELTA_ 1.0f
#define CHUNK 64
#define NCHUNK (NN / CHUNK)

typedef float v2f __attribute__((ext_vector_type(2)));
typedef float v8f __attribute__((ext_vector_type(8)));

// fp32 WMMA: D(16x16,f32) = A(16x4,f32) * B(4x16,f32) + C
__device__ __forceinline__ v8f wmma4(v2f a, v2f b, v8f c) {
  return __builtin_amdgcn_wmma_f32_16x16x4_f32(
      /*neg_a=*/false, a, /*neg_b=*/false, b,
      /*c_mod=*/(short)0, c, /*reuse_a=*/false, /*reuse_b=*/false);
}

// LDS byte offset of a (dynamic) shared-memory pointer (generic -> AS3 strip).
__device__ __forceinline__ unsigned lds_off(const void* p) {
  return (unsigned)(unsigned long long)(__attribute__((address_space(3))) const void*)p;
}

// Row-block i0 (16 rows) of P = A(128x128) * B(128x128), both row-major, into acc[8].
__device__ __forceinline__ void mm_rowblock(const float* __restrict__ A,
                                            const float* __restrict__ B,
                                            int i0, int m, int kb, v8f acc[8]) {
  for (int k = 0; k < DD; k += 4) {
    v2f a;
    a.x = A[(i0 + m) * DD + k + kb];
    a.y = A[(i0 + m) * DD + k + kb + 1];
#pragma unroll
    for (int j = 0; j < 8; ++j) {
      v2f b;
      b.x = B[(k + kb) * DD + j * 16 + m];
      b.y = B[(k + kb + 1) * DD + j * 16 + m];
      acc[j] = wmma4(a, b, acc[j]);
    }
  }
}

// Scatter D tiles (C/D layout: row = i0 + v + 8*(lane>=16), col = j*16 + (lane&15))
__device__ __forceinline__ void store_rowblock(float* __restrict__ Dst, int i0,
                                               int lane, v8f acc[8]) {
  const int m = lane & 15;
  const int rbase = i0 + ((lane >> 4) << 3);
#pragma unroll
  for (int j = 0; j < 8; ++j) {
#pragma unroll
    for (int v = 0; v < 8; ++v) {
      Dst[(rbase + v) * DD + j * 16 + m] = acc[j][v];
    }
  }
}

// -------------------------------------------------------------------------
// Kernel 1: per-class Gram matrices  G[c] = X_c^T X_c  (c = 64 -> unmasked)
// 65 blocks x 256 threads (8 waves). Double-buffered async global->LDS copy
// (GLOBAL_LOAD_ASYNC_TO_LDS_B128 / s_wait_asynccnt). Since the row mask M is
// diagonal idempotent, X^T M X == (MX)^T X, so only the A operand is masked.
// -------------------------------------------------------------------------
__global__ void gram_kernel(const float* __restrict__ feat,
                            const int* __restrict__ y,
                            float* __restrict__ G) {
  extern __shared__ float smem[];
  float* tile0 = smem;                              // 32 KB
  float* tile1 = smem + CHUNK * DD;                 // 32 KB
  float* mskb = smem + 2 * CHUNK * DD;              // 2 x 64 mask slots

  const int c = blockIdx.x;        // 0..63 classes, 64 == all rows
  const int tid = threadIdx.x;
  const int wave = tid >> 5;
  const int lane = tid & 31;
  const int m = lane & 15;
  const int kb = (lane >> 4) << 1; // lanes 0-15 -> K 0,1 ; lanes 16-31 -> K 2,3
  const int i0 = wave * 16;

  // Issue 8 async 16B copies per thread: one 64x128 f32 chunk -> LDS buffer.
  auto issue_chunk = [&](int ch, float* buf) {
#pragma unroll
    for (int q = 0; q < 8; ++q) {
      int idx = tid + q * 256;                       // float4 index within chunk
      const float4* gp = (const float4*)feat + (size_t)ch * (CHUNK * 32) + idx;
      unsigned lo = lds_off(buf) + (unsigned)idx * 16u;
      asm volatile("global_load_async_to_lds_b128 %0, %1, off"
                   :: "v"(lo), "v"(gp) : "memory");
    }
  };
  auto write_mask = [&](int ch, float* mb) {
    if (tid < CHUNK) {
      mb[tid] = (c == CC) ? 1.0f : ((y[ch * CHUNK + tid] == c) ? 1.0f : 0.0f);
    }
  };

  v8f acc[8] = {};

  write_mask(0, mskb);
  issue_chunk(0, tile0);

  for (int ch = 0; ch < NCHUNK; ++ch) {
    const float* cur = (ch & 1) ? tile1 : tile0;
    const float* cm = mskb + (ch & 1) * CHUNK;
    if (ch + 1 < NCHUNK) {
      write_mask(ch + 1, mskb + ((ch + 1) & 1) * CHUNK);
      issue_chunk(ch + 1, ((ch + 1) & 1) ? tile1 : tile0);
      asm volatile("s_wait_asynccnt 0x8" ::: "memory");  // current chunk landed
    } else {
      asm volatile("s_wait_asynccnt 0x0" ::: "memory");
    }
    __syncthreads();
    // G tile(i0, j) += (M X)^T X over this chunk (mask applied to A only).
    for (int k = 0; k < CHUNK; k += 4) {
      v2f a;                              // A-op = (MX)^T : A[m][kq] = msk*X[k+kq][i0+m]
      a.x = cur[(k + kb) * DD + i0 + m] * cm[k + kb];
      a.y = cur[(k + kb + 1) * DD + i0 + m] * cm[k + kb + 1];
#pragma unroll
      for (int j = 0; j < 8; ++j) {
        v2f b;                            // B-op = X : B[kq][n] = X[k+kq][j*16+n]
        b.x = cur[(k + kb) * DD + j * 16 + m];
        b.y = cur[(k + kb + 1) * DD + j * 16 + m];
        acc[j] = wmma4(a, b, acc[j]);
      }
    }
    __syncthreads();   // compute done before next iteration's async overwrites
  }

  store_rowblock(G + (size_t)c * DD * DD, i0, lane, acc);
}

// -------------------------------------------------------------------------
// Kernel 2: nuclear norm via Newton-Schulz sqrt, fully LDS-resident.
// A = G/trace(G); Y->sqrt(A), Z->sqrt(A)^-1; nuc = sqrt(trace(G))*trace(Y).
// 192 KB dynamic LDS (Y, Z, M) -- needs CDNA5's 320 KB WGP LDS.
// -------------------------------------------------------------------------
__global__ void ns_kernel(const float* __restrict__ G, float* __restrict__ nuc) {
  extern __shared__ float lds[];
  float* Y = lds;
  float* Z = lds + DD * DD;
  float* M = lds + 2 * DD * DD;

  const int c = blockIdx.x;
  const float* Gc = G + (size_t)c * DD * DD;
  const int tid = threadIdx.x;
  const int wave = tid >> 5;
  const int lane = tid & 31;
  const int m = lane & 15;
  const int kb = (lane >> 4) << 1;
  const int i0 = wave * 16;

  __shared__ float s_trace;
  if (tid < DD) M[tid] = Gc[tid * DD + tid];
  __syncthreads();
  if (tid == 0) {
    float s = 0.f;
    for (int i = 0; i < DD; ++i) s += M[i];
    s_trace = s;
  }
  __syncthreads();
  const float s = s_trace;
  if (s <= 1e-30f) {               // empty class -> ||X_c||_* = 0
    if (tid == 0) nuc[c] = 0.0f;
    return;
  }
  const float inv = 1.0f / s;

  // Y = G/s (spectrum in (0,1]), Z = I
  for (int q = 0; q < 64; ++q) {
    int idx = tid + q * 256;
    Y[idx] = Gc[idx] * inv;
    int r = idx >> 7, col = idx & 127;
    Z[idx] = (r == col) ? 1.0f : 0.0f;
  }
  __syncthreads();

  for (int it = 0; it < NS_ITERS; ++it) {
    // M = Z * Y
    {
      v8f t[8] = {};
      mm_rowblock(Z, Y, i0, m, kb, t);
      __syncthreads();
      store_rowblock(M, i0, lane, t);
      __syncthreads();
    }
    // M = 0.5 * (3I - M)
    for (int q = 0; q < 64; ++q) {
      int idx = tid + q * 256;
      int r = idx >> 7, col = idx & 127;
      M[idx] = 0.5f * (((r == col) ? 3.0f : 0.0f) - M[idx]);
    }
    __syncthreads();
    // Y' = Y*M ; Z' = M*Z   (held in registers, then written back)
    v8f ny[8] = {};
    v8f nz[8] = {};
    mm_rowblock(Y, M, i0, m, kb, ny);
    mm_rowblock(M, Z, i0, m, kb, nz);
    __syncthreads();
    store_rowblock(Y, i0, lane, ny);
    store_rowblock(Z, i0, lane, nz);
    __syncthreads();
  }

  // nuc = sqrt(s) * trace(Y)
  if (tid < DD) M[tid] = Y[tid * DD + tid];
  __syncthreads();
  if (tid == 0) {
    float t = 0.f;
    for (int i = 0; i < DD; ++i) t += M[i];
    nuc[c] = sqrtf(s) * t;
  }
}

// -------------------------------------------------------------------------
// Kernel 3: cross entropy partial sums (one row per thread, block reduce)
// -------------------------------------------------------------------------
__global__ void xe_kernel(const float* __restrict__ out,
                          const int* __restrict__ y,
                          float* __restrict__ part) {
  const int tid = threadIdx.x;
  const int r = blockIdx.x * blockDim.x + tid;  // 32 blocks * 256 = 8192
  const float* o = out + (size_t)r * CC;
  float mx = -3.402823466e38f;
  for (int j = 0; j < CC; ++j) mx = fmaxf(mx, o[j]);
  float se = 0.f;
  for (int j = 0; j < CC; ++j) se += __expf(o[j] - mx);
  float xe = __logf(se) + mx - o[y[r]];

  __shared__ float red[256];
  red[tid] = xe;
  __syncthreads();
  for (int off = 128; off > 0; off >>= 1) {
    if (tid < off) red[tid] += red[tid + off];
    __syncthreads();
  }
  if (tid == 0) part[blockIdx.x] = red[0];
}

// -------------------------------------------------------------------------
// Kernel 4: combine -> scalar loss
// -------------------------------------------------------------------------
__global__ void final_kernel(const float* __restrict__ nuc,
                             const float* __restrict__ part,
                             float* __restrict__ loss) {
  if (threadIdx.x == 0) {
    float obj = 0.f;
    for (int c = 0; c < CC; ++c) obj += fmaxf(nuc[c], DELTA_);
    float ole = (obj - nuc[CC]) * (LAMBDA_ / (float)NN);
    float xs = 0.f;
    for (int b = 0; b < 32; ++b) xs += part[b];
    loss[0] = ole + xs / (float)NN;
  }
}

extern "C" void kernel_launch(void* const* d_in, const int* in_sizes, int n_in,
                              void* d_out, int out_size, void* d_ws, size_t ws_size,
                              hipStream_t stream) {
  const float* out_logits = (const float*)d_in[0];  // [8192, 64] f32
  const float* feat       = (const float*)d_in[1];  // [8192, 128] f32
  const int*   y          = (const int*)d_in[2];    // [8192] i32
  float* loss = (float*)d_out;

  float* G    = (float*)d_ws;                       // 65 * 128*128 f32 (~4.25 MB)
  float* nuc  = G + (size_t)(CC + 1) * DD * DD;     // 65 f32
  float* part = nuc + (CC + 1);                     // 32 f32

  size_t gram_lds = (2 * CHUNK * DD + 2 * CHUNK) * sizeof(float);  // ~66 KB
  gram_kernel<<<CC + 1, 256, gram_lds, stream>>>(feat, y, G);
  ns_kernel<<<CC + 1, 256, 3 * DD * DD * sizeof(float), stream>>>(G, nuc);
  xe_kernel<<<NN / 256, 256, 0, stream>>>(out_logits, y, part);
  final_kernel<<<1, 64, 0, stream>>>(nuc, part, loss);
}